// TensorTrainEmbedding_54245436949057
// MI455X (gfx1250) — compile-verified
//
#include <hip/hip_runtime.h>

typedef __attribute__((ext_vector_type(2))) float v2f;
typedef __attribute__((ext_vector_type(4))) float f4;
typedef __attribute__((ext_vector_type(8))) float v8f;

#define WAVES_PER_BLOCK 4

// Tensor-train embedding, DIM=64, N_CHUNKS=3, D=4, RANK=16.
// One batch element per wave32. Gather rows coalesced into LDS, then do the
// per-element contraction chain with V_WMMA_F32_16X16X4_F32 (K=16 via 4 chained ops).
__global__ __launch_bounds__(WAVES_PER_BLOCK * 32)
void tt_embed_kernel(const int* __restrict__ hs,
                     const float* __restrict__ start_core,
                     const float* __restrict__ end_core,
                     const float* __restrict__ cores,
                     float* __restrict__ out,
                     int n_batch)
{
    // Per-wave LDS: core row (1024) | v0 padded 16x16 (256) | start padded 16x16 (256) | M 16x16 (256)
    __shared__ float lds[WAVES_PER_BLOCK][1792];

    const int lane = threadIdx.x & 31;
    const int wave = threadIdx.x >> 5;
    const int b    = blockIdx.x * WAVES_PER_BLOCK + wave;
    if (b >= n_batch) return;   // uniform per wave -> whole wave exits together

    const int h0 = hs[b * 3 + 0];
    const int h1 = hs[b * 3 + 1];
    const int h2 = hs[b * 3 + 2];

    float* L  = lds[wave];
    float* Lc = L;            // core[d][s][r] : 4*16*16
    float* Lv = L + 1024;     // v0 padded to 16x16, rows e>=4 zero (row e, col r)
    float* Ls = L + 1280;     // start padded to 16x16, rows d'>=4 zero (row d', col r)
    float* LM = L + 1536;     // M (row d*4+e, col s)

    // ---- coalesced gather: 4KB core row as 8 x b128 per lane ----
    const f4* gc = (const f4*)(cores + (size_t)h1 * 1024);
    f4* lc4 = (f4*)Lc;
#pragma unroll
    for (int i = 0; i < 8; ++i)
        lc4[i * 32 + lane] = gc[i * 32 + lane];

    // ---- end/start rows (64 floats each), zero-padded to 16x16 in LDS ----
    const float* ge = end_core   + (size_t)h2 * 64;
    const float* gs = start_core + (size_t)h0 * 64;
#pragma unroll
    for (int i = 0; i < 8; ++i) {
        int idx = i * 32 + lane;
        Lv[idx] = (idx < 64) ? ge[idx] : 0.0f;
        Ls[idx] = (idx < 64) ? gs[idx] : 0.0f;
    }

    // f32 WMMA operand indexing (wave32):
    //   A 16x4 chunk: lane m = lane%16; VGPR0/1 hold K = {0,1} (lanes 0-15) or {2,3} (lanes 16-31)
    //   B 4x16 chunk: lane n = lane%16; same K split (symmetric to A)
    //   C/D 16x16:    VGPR j holds (m = j + 8*(lane>=16), n = lane%16)
    const int m     = lane & 15;
    const int khalf = (lane >> 4) << 1;   // 0 or 2
    const int srow0 = (lane >> 4) << 3;   // 0 or 8

    // ---- Stage 1: Dd[s,e] = sum_r core[d][s][r] * v0[e][r]   (B = v0^T, cols e>=4 zero) ----
#pragma unroll
    for (int d = 0; d < 4; ++d) {
        v8f acc = {};
#pragma unroll
        for (int c = 0; c < 4; ++c) {
            const int k = 4 * c + khalf;
            v2f A;  A.x  = Lc[d * 256 + m * 16 + k];  A.y  = Lc[d * 256 + m * 16 + k + 1];
            v2f Bv; Bv.x = Lv[m * 16 + k];            Bv.y = Lv[m * 16 + k + 1];
            acc = __builtin_amdgcn_wmma_f32_16x16x4_f32(
                      false, A, false, Bv, (short)0, acc, false, false);
        }
        // scatter valid columns (e = n < 4) into M[(d*4+e), s] rows
        if (m < 4) {
#pragma unroll
            for (int j = 0; j < 8; ++j)
                LM[(d * 4 + m) * 16 + (srow0 + j)] = acc[j];
        }
    }

    // ---- Stage 2: out[d',e'] = sum_s start[d'][s] * M[e'][s]  (A = padded start, B[k=s,n=e']=M[e',s]) ----
    v8f acc2 = {};
#pragma unroll
    for (int c = 0; c < 4; ++c) {
        const int k = 4 * c + khalf;
        v2f A;  A.x  = Ls[m * 16 + k];  A.y  = Ls[m * 16 + k + 1];
        v2f Bv; Bv.x = LM[m * 16 + k];  Bv.y = LM[m * 16 + k + 1];
        acc2 = __builtin_amdgcn_wmma_f32_16x16x4_f32(
                   false, A, false, Bv, (short)0, acc2, false, false);
    }

    // valid output rows m_row = j < 4 live in lanes 0-15, VGPRs 0-3
    if (lane < 16) {
        float* ob = out + (size_t)b * 64;
#pragma unroll
        for (int j = 0; j < 4; ++j)
            ob[j * 16 + lane] = acc2[j];
    }
}

extern "C" void kernel_launch(void* const* d_in, const int* in_sizes, int n_in,
                              void* d_out, int out_size, void* d_ws, size_t ws_size,
                              hipStream_t stream) {
    const int*   hs         = (const int*)  d_in[0];
    const float* start_core = (const float*)d_in[1];
    const float* end_core   = (const float*)d_in[2];
    const float* cores      = (const float*)d_in[3];
    float*       out        = (float*)d_out;

    const int n_batch = in_sizes[0] / 3;                 // hs is (B, 3)
    const int blocks  = (n_batch + WAVES_PER_BLOCK - 1) / WAVES_PER_BLOCK;
    tt_embed_kernel<<<blocks, WAVES_PER_BLOCK * 32, 0, stream>>>(
        hs, start_core, end_core, cores, out, n_batch);
}